// LengthRegulator_32555852104214
// MI455X (gfx1250) — compile-verified
//
#include <hip/hip_runtime.h>
#include <stdint.h>

// ---------------------------------------------------------------------------
// Kernel 1: per-batch inclusive scan of durations (S <= 1024, one block/batch)
// Writes cum[B,S] to workspace and out_lens[b] (as float) to the d_out tail.
// ---------------------------------------------------------------------------
__global__ void lr_scan_kernel(const int* __restrict__ dur,
                               int* __restrict__ cum,
                               float* __restrict__ lens_out,
                               int S) {
    __shared__ int s[1024];
    const int b = blockIdx.x;
    const int t = threadIdx.x;
    int v = (t < S) ? dur[b * S + t] : 0;
    s[t] = v;
    __syncthreads();
#pragma unroll
    for (int off = 1; off < 1024; off <<= 1) {
        int add = (t >= off) ? s[t - off] : 0;
        __syncthreads();
        s[t] += add;
        __syncthreads();
    }
    if (t < S) cum[b * S + t] = s[t];
    if (t == S - 1) lens_out[b] = (float)s[t];
}

// ---------------------------------------------------------------------------
// Kernel 2: one wave32 per output row (b, j).
//   - binary search cum[b,:] for first t with cum[t] > j (searchsorted right)
//   - copy x[b,t,:] (D=384 floats = 1536B) to out row via CDNA5 async
//     global->LDS->global 128b transfers (512B per wave per instruction)
//   - rows j >= out_lens[b] are zero-filled
// blockDim must be 256 (8 waves); D must be 384.
// ---------------------------------------------------------------------------
__global__ void lr_expand_kernel(const float* __restrict__ x,
                                 const int* __restrict__ cum,
                                 float* __restrict__ out,
                                 int S, int L, int nrows) {
    constexpr int D = 384;                 // floats per row
    constexpr int ITERS = D / 128;         // 3: 32 lanes x 16B x 3 = 1536B
    __shared__ __align__(16) float stage[8 * D];

    const int lane = threadIdx.x & 31;
    const int wave = threadIdx.x >> 5;
    const int row  = blockIdx.x * 8 + wave;
    if (row >= nrows) return;

    const int b = row / L;
    const int j = row - b * L;
    const int* __restrict__ c = cum + (size_t)b * S;
    float* __restrict__ dst = out + (size_t)row * D;

    const int total = c[S - 1];
    if (j >= total) {
        // padded tail: zero-initialized output
        const float4 z = {0.f, 0.f, 0.f, 0.f};
#pragma unroll
        for (int i = 0; i < ITERS; ++i)
            reinterpret_cast<float4*>(dst)[i * 32 + lane] = z;
        return;
    }

    // searchsorted(cum, j, side='right'), clipped to S-1 (uniform per wave)
    int lo = 0, hi = S - 1;
    while (lo < hi) {
        int mid = (lo + hi) >> 1;
        if (c[mid] > j) hi = mid; else lo = mid + 1;
    }
    const float* __restrict__ src = x + ((size_t)b * S + lo) * (size_t)D;

    // per-wave LDS staging slot (1536B); LDS byte offset = low 32 bits of the
    // flat shared-aperture address (hardware truncates addr[31:0] into LDS)
    unsigned lds_base = (unsigned)(uintptr_t)(void*)&stage[wave * D];

#pragma unroll
    for (int i = 0; i < ITERS; ++i) {
        const float* g = src + (i * 32 + lane) * 4;
        unsigned     l = lds_base + (unsigned)((i * 32 + lane) * 16);
        asm volatile("global_load_async_to_lds_b128 %0, %1, off"
                     :: "v"(l), "v"(g) : "memory");
    }
    asm volatile("s_wait_asynccnt 0x0" ::: "memory");
#pragma unroll
    for (int i = 0; i < ITERS; ++i) {
        float*   g = dst + (i * 32 + lane) * 4;
        unsigned l = lds_base + (unsigned)((i * 32 + lane) * 16);
        asm volatile("global_store_async_from_lds_b128 %0, %1, off"
                     :: "v"(g), "v"(l) : "memory");
    }
    asm volatile("s_wait_asynccnt 0x0" ::: "memory");
}

// ---------------------------------------------------------------------------
// Launch: L is recovered host-side from out_size = B*L*D + B (no device sync,
// graph-capture safe). Workspace holds cum[B,S] (128 KB).
// ---------------------------------------------------------------------------
extern "C" void kernel_launch(void* const* d_in, const int* in_sizes, int n_in,
                              void* d_out, int out_size, void* d_ws, size_t ws_size,
                              hipStream_t stream) {
    const float* x   = (const float*)d_in[0];
    const int*   dur = (const int*)d_in[1];

    const int B  = 32;                    // fixed by reference setup
    const int BS = in_sizes[1];           // B*S
    const int S  = BS / B;                // 1024
    const int D  = in_sizes[0] / BS;      // 384
    const int L  = (out_size - B) / (B * D);

    float* out  = (float*)d_out;
    float* lens = out + (size_t)B * (size_t)L * (size_t)D;  // tail: out_lens
    int*   cum  = (int*)d_ws;                               // B*S ints

    lr_scan_kernel<<<B, 1024, 0, stream>>>(dur, cum, lens, S);

    const int nrows  = B * L;             // one wave per row
    const int blocks = (nrows + 7) / 8;   // 8 waves (256 threads) per block
    lr_expand_kernel<<<blocks, 256, 0, stream>>>(x, cum, out, S, L, nrows);
}